// Agent2Lane_emb_proj_30537217474893
// MI455X (gfx1250) — compile-verified
//
#include <hip/hip_runtime.h>

typedef __attribute__((ext_vector_type(2))) float v2f;
typedef __attribute__((ext_vector_type(8))) float v8f;

#define NT_TOT   1024   // N*T
#define M_LANES  64
#define L_PTS    16
#define NEMBD    128
#define EDIM     12
#define INV_SCALE 0.1f

__global__ __launch_bounds__(256) void a2l_edge_proj_wmma(
    const float* __restrict__ agent,   // (8,1024,8)
    const float* __restrict__ lane,    // (8,64,64)
    const float* __restrict__ W,       // (128,12)
    const float* __restrict__ bias,    // (128)
    float* __restrict__ out)           // (8,1024,64,128)
{
    __shared__ float sW[NEMBD * EDIM];
    __shared__ float sB[NEMBD];
    const int tid = threadIdx.x;
    for (int i = tid; i < NEMBD * EDIM; i += 256) sW[i] = W[i];
    if (tid < NEMBD) sB[tid] = bias[tid];
    __syncthreads();

    const int wave = tid >> 5;
    const int l    = tid & 31;
    const int l15  = l & 15;
    const bool lo  = (l < 16);

    const int tile = blockIdx.x * 8 + wave;   // 0..32767, exactly covered
    const int bnt  = tile >> 2;               // b*1024 + nt
    const int b    = bnt >> 10;
    const int m    = ((tile & 3) << 4) + l15; // this lane's edge row's m
    const long long row_base = (long long)tile * 16;

    // ---- agent row (wave-uniform) ----
    const float* ag = agent + (size_t)bnt * 8;
    float4 a0 = *(const float4*)(ag);
    float4 a1 = *(const float4*)(ag + 4);
    const float px = a0.x, py = a0.y, ps = a0.w, pc = a1.x;
    const bool z1 = (a0.x==0.f)&&(a0.y==0.f)&&(a0.z==0.f)&&(a0.w==0.f)&&
                    (a1.x==0.f)&&(a1.y==0.f)&&(a1.z==0.f)&&(a1.w==0.f);
    const float f1 = z1 ? 0.f : 1.f;

    // ---- per-row 12-dim edge feature (lanes l and l+16 compute same row) ----
    float eg[12];
    const float* lp = lane + ((size_t)(b * M_LANES + m)) * 64;
    float best = __builtin_inff();
    float e0 = 0.f, e1 = 0.f, e2 = 0.f, e3 = 0.f;
#pragma unroll
    for (int p = 0; p < L_PTS; ++p) {
        float4 lv = *(const float4*)(lp + p * 4);
        const float lx = lv.x, ly = lv.y, ls = lv.z, lc = lv.w;
        const bool z2 = (lx==0.f)&&(ly==0.f)&&(ls==0.f)&&(lc==0.f);
        const float f = z2 ? 0.f : f1;
        const float dxw = px - lx, dyw = py - ly;
        const float d0 = ( dxw * lc + dyw * ls) * INV_SCALE * f;
        const float d1 = (-dxw * ls + dyw * lc) * INV_SCALE * f;
        const float d2 = (ps * lc - pc * ls) * f;
        const float d3 = (pc * lc + ps * ls) * f;
        const float ad = fabsf(d0);
        if (ad < best) { best = ad; e0 = d0; e1 = d1; e2 = d2; e3 = d3; }
        if (p == 0)         { eg[4] = d0; eg[5] = d1; eg[6]  = d2; eg[7]  = d3; }
        if (p == L_PTS - 1) { eg[8] = d0; eg[9] = d1; eg[10] = d2; eg[11] = d3; }
    }
    eg[0] = e0; eg[1] = e1; eg[2] = e2; eg[3] = e3;

    // ---- WMMA: D(16x16) = A(16x4) * B(4x16) + C, 3 k-steps x 8 col tiles ----
    float* outp = out + (size_t)row_base * NEMBD;
    const int rofs = lo ? 0 : 8;

#pragma unroll
    for (int c = 0; c < 8; ++c) {
        const int e = c * 16 + l15;            // output column this lane feeds
        const float bb = sB[e];
        v8f acc = {bb, bb, bb, bb, bb, bb, bb, bb};   // bias folded into C
#pragma unroll
        for (int s = 0; s < 3; ++s) {
            v2f A, B;
            // A layout 16x4 f32: lanes 0-15 -> K = 4s+0,4s+1 ; lanes 16-31 -> K = 4s+2,4s+3
            A.x = lo ? eg[4*s + 0] : eg[4*s + 2];
            A.y = lo ? eg[4*s + 1] : eg[4*s + 3];
            // B[k][n] = W[n][k] (einsum 'bnmd,ed->bnme'): same half-split over K
            const int wofs = e * EDIM + 4*s + (lo ? 0 : 2);
            B.x = sW[wofs];
            B.y = sW[wofs + 1];
            acc = __builtin_amdgcn_wmma_f32_16x16x4_f32(
                false, A, false, B, (short)0, acc, false, false);
        }
        // D layout: VGPR v -> row v (lanes 0-15) / row v+8 (lanes 16-31), col e
#pragma unroll
        for (int v = 0; v < 8; ++v) {
            outp[(size_t)(v + rofs) * NEMBD + e] = acc[v];
        }
    }
}

extern "C" void kernel_launch(void* const* d_in, const int* in_sizes, int n_in,
                              void* d_out, int out_size, void* d_ws, size_t ws_size,
                              hipStream_t stream) {
    (void)in_sizes; (void)n_in; (void)out_size; (void)d_ws; (void)ws_size;
    const float* agent = (const float*)d_in[0];
    const float* lane  = (const float*)d_in[1];
    const float* W     = (const float*)d_in[2];
    const float* bias  = (const float*)d_in[3];
    float* out = (float*)d_out;
    // 32768 row-tiles of 16 rows, 8 waves (256 threads) per block -> 4096 blocks
    a2l_edge_proj_wmma<<<4096, 256, 0, stream>>>(agent, lane, W, bias, out);
}